// SimpleSampling_87866440942237
// MI455X (gfx1250) — compile-verified
//
#include <hip/hip_runtime.h>

typedef float v2f __attribute__((ext_vector_type(2)));
typedef float v4f __attribute__((ext_vector_type(4)));
typedef float v8f __attribute__((ext_vector_type(8)));

#define BDIM   8192
#define DDIM   128
#define TOPK   10
#define CHUNK  1024
#define CHUNKP (CHUNK + 1)           // pad: (r*1025+c)%64 = (r+c)%64 -> conflict-free scan
#define MWPR   (BDIM / 32)           // mask words per row = 256

// ---------------------------------------------------------------------------
// Phase 0: build exclusion bitmask (diagonal + social edges)
// ---------------------------------------------------------------------------
__global__ void mask_init_kernel(const int* __restrict__ edges, int E,
                                 unsigned* __restrict__ mask) {
  int t = blockIdx.x * blockDim.x + threadIdx.x;
  if (t < BDIM) {
    atomicOr(&mask[(size_t)t * MWPR + (t >> 5)], 1u << (t & 31));
  }
  if (t < E) {
    int s = edges[t];
    int d = edges[E + t];
    atomicOr(&mask[(size_t)s * MWPR + (d >> 5)], 1u << (d & 31));
  }
}

// ---------------------------------------------------------------------------
// Phase 1: proj = user_emb @ W^T + b  via V_WMMA_F32_16X16X4_F32
// One wave per 16x16 output tile; K=128 -> 32 WMMA steps.
// A lane layout (16x4 f32): M = lane%16, K = {koff, koff+1}, koff = 2*(lane>=16)
// ---------------------------------------------------------------------------
__global__ void proj_kernel(const float* __restrict__ U, const float* __restrict__ W,
                            const float* __restrict__ bias, float* __restrict__ P) {
  int bid  = blockIdx.x;               // 4096 tiles: 512 row-tiles x 8 col-tiles
  int m0   = (bid >> 3) << 4;
  int n0   = (bid & 7) << 4;
  int l    = threadIdx.x;
  int lr   = l & 15;
  int koff = (l >> 4) << 1;

  const float* arow = U + (size_t)(m0 + lr) * DDIM + koff;  // user_emb row
  const float* brow = W + (size_t)(n0 + lr) * DDIM + koff;  // B[k][n] = W[n][k]

  v8f c = {0.f, 0.f, 0.f, 0.f, 0.f, 0.f, 0.f, 0.f};
#pragma unroll
  for (int s = 0; s < 32; ++s) {
    v2f a = *(const v2f*)(arow + 4 * s);
    v2f b = *(const v2f*)(brow + 4 * s);
    c = __builtin_amdgcn_wmma_f32_16x16x4_f32(false, a, false, b, (short)0, c,
                                              false, false);
  }
  float bb    = bias[n0 + lr];
  int   rbase = m0 + ((l >> 4) << 3);  // C layout: VGPR v -> rows v / v+8
#pragma unroll
  for (int v = 0; v < 8; ++v) {
    P[(size_t)(rbase + v) * DDIM + n0 + lr] = c[v] + bb;
  }
}

// ---------------------------------------------------------------------------
// Phase 2 (fused): sim tiles (WMMA f32, 16x64 slab per wave, A reused 4x)
//   -> streaming top-10 per row -> single-pass non-temporal binary write.
// One workgroup (8 waves) per 16-row block; 1024-column chunks staged in LDS.
// Mask chunk staged via async global->LDS loads (ASYNCcnt path).
// ---------------------------------------------------------------------------
__global__ __launch_bounds__(256) void sim_topk_kernel(
    const float* __restrict__ P, const unsigned* __restrict__ mask,
    float* __restrict__ out) {
  __shared__ float    simChunk[16][CHUNKP];     // ~65.6 KB
  __shared__ unsigned maskCh[16][CHUNK / 32];   // 2 KB
  __shared__ float    topVal[16][TOPK];
  __shared__ int      topIdx[16][TOPK];

  int tid  = threadIdx.x;
  int l    = tid & 31;
  int wv   = tid >> 5;            // wave id 0..7 (uniform per wave)
  int lr   = l & 15;
  int koff = (l >> 4) << 1;
  int m0   = blockIdx.x << 4;

  const float* arow = P + (size_t)(m0 + lr) * DDIM + koff;  // shared A rows

  if (tid < 16 * TOPK) {
    topVal[tid / TOPK][tid % TOPK] = -3.0e38f;
    topIdx[tid / TOPK][tid % TOPK] = -1;
  }
  __syncthreads();

  for (int c0 = 0; c0 < BDIM; c0 += CHUNK) {
    // Stage this chunk's mask slice (16 rows x 32 words) with async->LDS loads.
    for (int i = tid; i < 16 * (CHUNK / 32); i += 256) {
      int r = i >> 5, w = i & 31;
      unsigned        ldsa = (unsigned)(unsigned long long)&maskCh[r][w];
      const unsigned* g    = &mask[(size_t)(m0 + r) * MWPR + (c0 >> 5) + w];
      asm volatile("global_load_async_to_lds_b32 %0, %1, off"
                   :
                   : "v"(ldsa), "v"(g)
                   : "memory");
    }
    asm volatile("s_wait_asynccnt 0x0" ::: "memory");

    // Compute 16x64 slabs: 4 column tiles per A load (A:B traffic = 1:4).
    for (int g = wv; g < CHUNK / 64; g += 8) {
      int          n0 = c0 + (g << 6);
      const float* b0 = P + (size_t)(n0 +  0 + lr) * DDIM + koff;
      const float* b1 = P + (size_t)(n0 + 16 + lr) * DDIM + koff;
      const float* b2 = P + (size_t)(n0 + 32 + lr) * DDIM + koff;
      const float* b3 = P + (size_t)(n0 + 48 + lr) * DDIM + koff;
      v8f acc0 = {0.f,0.f,0.f,0.f,0.f,0.f,0.f,0.f};
      v8f acc1 = acc0, acc2 = acc0, acc3 = acc0;
#pragma unroll
      for (int s = 0; s < 32; ++s) {
        v2f a   = *(const v2f*)(arow + 4 * s);
        v2f vb0 = *(const v2f*)(b0 + 4 * s);
        v2f vb1 = *(const v2f*)(b1 + 4 * s);
        v2f vb2 = *(const v2f*)(b2 + 4 * s);
        v2f vb3 = *(const v2f*)(b3 + 4 * s);
        acc0 = __builtin_amdgcn_wmma_f32_16x16x4_f32(false, a, false, vb0, (short)0, acc0, false, false);
        acc1 = __builtin_amdgcn_wmma_f32_16x16x4_f32(false, a, false, vb1, (short)0, acc1, false, false);
        acc2 = __builtin_amdgcn_wmma_f32_16x16x4_f32(false, a, false, vb2, (short)0, acc2, false, false);
        acc3 = __builtin_amdgcn_wmma_f32_16x16x4_f32(false, a, false, vb3, (short)0, acc3, false, false);
      }
      int rb = (l >> 4) << 3;
      int cb = (g << 6) + lr;
#pragma unroll
      for (int v = 0; v < 8; ++v) {
        simChunk[rb + v][cb +  0] = acc0[v];
        simChunk[rb + v][cb + 16] = acc1[v];
        simChunk[rb + v][cb + 32] = acc2[v];
        simChunk[rb + v][cb + 48] = acc3[v];
      }
    }
    __syncthreads();

    // Streaming top-10: one scanner thread per row (rare insert path).
    if (tid < 16) {
      int   r  = tid;
      float mt = topVal[r][TOPK - 1];
      for (int c = 0; c < CHUNK; ++c) {
        float v = simChunk[r][c];
        if (v > mt) {
          if (!((maskCh[r][c >> 5] >> (c & 31)) & 1u)) {
            int gidx = c0 + c;
            int p    = TOPK - 1;
            while (p > 0 && v > topVal[r][p - 1]) {  // strict: ties keep lower idx
              topVal[r][p] = topVal[r][p - 1];
              topIdx[r][p] = topIdx[r][p - 1];
              --p;
            }
            topVal[r][p] = v;
            topIdx[r][p] = gidx;
            mt           = topVal[r][TOPK - 1];
          }
        }
      }
    }
    __syncthreads();
  }

  // Single-pass binary write: 16 threads/row, 512 cols each, NT v4f stores
  // (keeps proj+mask resident in the 192MB L2; output streams to HBM).
  int r  = tid >> 4;
  int c0 = (tid & 15) << 9;
  int gi[TOPK];
#pragma unroll
  for (int p = 0; p < TOPK; ++p) gi[p] = topIdx[r][p];
  float* orow = out + (size_t)(m0 + r) * BDIM;
  for (int c = c0; c < c0 + 512; c += 4) {
    float f0 = 0.f, f1 = 0.f, f2 = 0.f, f3 = 0.f;
#pragma unroll
    for (int p = 0; p < TOPK; ++p) {
      int ip = gi[p];
      if (ip == c)     f0 = 1.f;
      if (ip == c + 1) f1 = 1.f;
      if (ip == c + 2) f2 = 1.f;
      if (ip == c + 3) f3 = 1.f;
    }
    v4f w = {f0, f1, f2, f3};
    __builtin_nontemporal_store(w, (v4f*)(orow + c));
  }
}

// ---------------------------------------------------------------------------
extern "C" void kernel_launch(void* const* d_in, const int* in_sizes, int n_in,
                              void* d_out, int out_size, void* d_ws, size_t ws_size,
                              hipStream_t stream) {
  const float* U     = (const float*)d_in[0];
  const int*   edges = (const int*)d_in[1];
  const float* W     = (const float*)d_in[2];
  const float* bias  = (const float*)d_in[3];
  float*       out   = (float*)d_out;

  float*    proj = (float*)d_ws;                                         // 4 MB
  unsigned* mask = (unsigned*)((char*)d_ws + (size_t)BDIM * DDIM * 4);   // 8 MB
  int       E    = in_sizes[1] / 2;

  (void)hipMemsetAsync(mask, 0, (size_t)BDIM * MWPR * sizeof(unsigned), stream);

  int nmax = (E > BDIM) ? E : BDIM;
  mask_init_kernel<<<(nmax + 255) / 256, 256, 0, stream>>>(edges, E, mask);

  proj_kernel<<<(BDIM / 16) * (DDIM / 16), 32, 0, stream>>>(U, W, bias, proj);

  sim_topk_kernel<<<BDIM / 16, 256, 0, stream>>>(proj, mask, out);
}